// FeatDP_66563403154015
// MI455X (gfx1250) — compile-verified
//
#include <hip/hip_runtime.h>
#include <cstdint>

// Problem dims (fixed by the reference)
#define N_NODES 8192
#define P_PATHS 3
#define IN_F    256
#define HID     64
#define HEADS   8
#define OUT_F   64
#define EDGES   262144
#define HD      512   // HEADS*HID

typedef __attribute__((ext_vector_type(2))) float v2f;
typedef __attribute__((ext_vector_type(8))) float v8f;

// fp32 WMMA: D(16x16,f32) = A(16x4,f32) * B(4x16,f32) + C
__device__ __forceinline__ v8f wmma4(v2f a, v2f b, v8f c) {
    return __builtin_amdgcn_wmma_f32_16x16x4_f32(
        /*neg_a=*/false, a, /*neg_b=*/false, b,
        /*c_mod=*/(short)0, c, /*reuse_a=*/false, /*reuse_b=*/false);
}

// ---------------- utility fills ----------------
__global__ void k_fill_f32(float* __restrict__ p, float v, int n) {
    int i = blockIdx.x * blockDim.x + threadIdx.x;
    if (i < n) p[i] = v;
}
__global__ void k_fill_u32(unsigned* __restrict__ p, unsigned v, int n) {
    int i = blockIdx.x * blockDim.x + threadIdx.x;
    if (i < n) p[i] = v;
}

// ---------------- feat = h @ fc[p]  (fp32 WMMA) ----------------
// grid: (HD/64, N/16, P), block: 32 (one wave -> 16 rows x 64 cols)
__global__ void k_gemm_feat(const float* __restrict__ h,
                            const float* __restrict__ fc,   // [P, IN, HD]
                            float* __restrict__ feat)       // [P, N, HD]
{
    const int lane = threadIdx.x;
    const int lo = lane & 15, hi = lane >> 4;
    const int col0 = blockIdx.x * 64;
    const int row0 = blockIdx.y * 16;
    const int p    = blockIdx.z;

    const float* __restrict__ arow = h + (size_t)(row0 + lo) * IN_F;
    const float* __restrict__ fcb  = fc + (size_t)p * IN_F * HD;

    v8f acc0 = {}, acc1 = {}, acc2 = {}, acc3 = {};
    for (int k = 0; k < IN_F; k += 4) {
        const int ka = k + 2 * hi;
        v2f a; a.x = arow[ka]; a.y = arow[ka + 1];
        const float* __restrict__ b0 = fcb + (size_t)ka * HD + col0 + lo;
        const float* __restrict__ b1 = b0 + HD;
        v2f b;
        b.x = b0[0];  b.y = b1[0];  acc0 = wmma4(a, b, acc0);
        b.x = b0[16]; b.y = b1[16]; acc1 = wmma4(a, b, acc1);
        b.x = b0[32]; b.y = b1[32]; acc2 = wmma4(a, b, acc2);
        b.x = b0[48]; b.y = b1[48]; acc3 = wmma4(a, b, acc3);
    }
    float* __restrict__ out = feat + ((size_t)p * N_NODES + row0) * HD + col0;
    #pragma unroll
    for (int r = 0; r < 8; ++r) {
        const size_t ro = (size_t)(r + 8 * hi) * HD + lo;
        out[ro +  0] = acc0[r];
        out[ro + 16] = acc1[r];
        out[ro + 32] = acc2[r];
        out[ro + 48] = acc3[r];
    }
}

// ---------------- el/er per (p, node, head) ----------------
__global__ void k_el_er(const float* __restrict__ feat,
                        const float* __restrict__ al,  // [P, HD]
                        const float* __restrict__ ar,
                        float* __restrict__ el, float* __restrict__ er)
{
    int idx = blockIdx.x * blockDim.x + threadIdx.x;     // P*N*HEADS
    if (idx >= P_PATHS * N_NODES * HEADS) return;
    const int hh = idx & 7;
    const int n  = (idx >> 3) % N_NODES;
    const int p  = idx / (N_NODES * HEADS);
    const float* __restrict__ f  = feat + ((size_t)p * N_NODES + n) * HD + hh * HID;
    const float* __restrict__ wl = al + (size_t)p * HD + hh * HID;
    const float* __restrict__ wr = ar + (size_t)p * HD + hh * HID;
    float sl = 0.f, sr = 0.f;
    #pragma unroll 8
    for (int d = 0; d < HID; ++d) { float fv = f[d]; sl += fv * wl[d]; sr += fv * wr[d]; }
    el[idx] = sl; er[idx] = sr;
}

// ---------------- edge logits + segment max (bit-trick float max) ----------------
__global__ void k_edge_logits(const int* __restrict__ srcl, const int* __restrict__ dstl,
                              const float* __restrict__ el, const float* __restrict__ er,
                              float* __restrict__ ebuf, unsigned* __restrict__ emax)
{
    int idx = blockIdx.x * blockDim.x + threadIdx.x;     // P*E*HEADS
    if (idx >= P_PATHS * EDGES * HEADS) return;
    const int hh = idx & 7;
    const int pe = idx >> 3;                // p*E + e
    const int p  = pe / EDGES;
    const int s  = srcl[pe], d = dstl[pe];
    float v = el[((size_t)p * N_NODES + s) * HEADS + hh] +
              er[((size_t)p * N_NODES + d) * HEADS + hh];
    v = v > 0.f ? v : 0.2f * v;             // leaky_relu(0.2)
    ebuf[idx] = v;
    unsigned* addr = &emax[((size_t)p * N_NODES + d) * HEADS + hh];
    if (v >= 0.f) atomicMax((int*)addr, __float_as_int(v));
    else          atomicMin(addr, __float_as_uint(v));
}

// ---------------- exp(e - max) + segment sum ----------------
__global__ void k_edge_exp(const int* __restrict__ dstl,
                           const unsigned* __restrict__ emax,
                           float* __restrict__ ebuf, float* __restrict__ denom)
{
    int idx = blockIdx.x * blockDim.x + threadIdx.x;     // P*E*HEADS
    if (idx >= P_PATHS * EDGES * HEADS) return;
    const int hh = idx & 7;
    const int pe = idx >> 3;
    const int p  = pe / EDGES;
    const int d  = dstl[pe];
    const float m  = __uint_as_float(emax[((size_t)p * N_NODES + d) * HEADS + hh]);
    const float ee = __expf(ebuf[idx] - m);
    ebuf[idx] = ee;
    atomicAdd(&denom[((size_t)p * N_NODES + d) * HEADS + hh], ee);
}

// ---------------- a = ee/denom; rst[dst] += a*feat[src]; alpha mean ----------------
// one wave per edge; each lane handles 16 contiguous channels (head = lane>>2)
__global__ void k_edge_scatter(const int* __restrict__ srcl, const int* __restrict__ dstl,
                               const float* __restrict__ ebuf, const float* __restrict__ denom,
                               const float* __restrict__ feat, float* __restrict__ rst,
                               float* __restrict__ amean)
{
    const int wid  = (blockIdx.x * blockDim.x + threadIdx.x) >> 5;  // p*E + e
    const int lane = threadIdx.x & 31;
    if (wid >= P_PATHS * EDGES) return;
    const int p = wid / EDGES;
    const int s = srcl[wid], d = dstl[wid];
    const int hh = lane >> 2;
    const float a = ebuf[(size_t)wid * HEADS + hh] /
                    denom[((size_t)p * N_NODES + d) * HEADS + hh];
    const float* __restrict__ f = feat + ((size_t)p * N_NODES + s) * HD + lane * 16;
    float* __restrict__ r       = rst  + ((size_t)p * N_NODES + d) * HD + lane * 16;
    #pragma unroll
    for (int j = 0; j < 16; ++j) atomicAdd(&r[j], a * f[j]);
    if (lane == 0) {
        float sa = 0.f;
        #pragma unroll
        for (int q = 0; q < HEADS; ++q)
            sa += ebuf[(size_t)wid * HEADS + q] /
                  denom[((size_t)p * N_NODES + d) * HEADS + q];
        amean[wid] = sa * (1.f / HEADS);
    }
}

// ---------------- zp = elu(rst + bias), in place ----------------
__global__ void k_bias_elu(float* __restrict__ rst, const float* __restrict__ bias)
{
    int idx = blockIdx.x * blockDim.x + threadIdx.x;     // P*N*HD
    if (idx >= P_PATHS * N_NODES * HD) return;
    const int hd = idx % HD;
    const int p  = idx / (N_NODES * HD);
    const float x = rst[idx] + bias[p * HD + hd];
    rst[idx] = x > 0.f ? x : __expf(x) - 1.f;
}

// ---------------- semantic attn: v = W1 @ w2, c = b1 . w2 ----------------
__global__ void k_sem_v(const float* __restrict__ w1, const float* __restrict__ b1,
                        const float* __restrict__ w2,
                        float* __restrict__ v, float* __restrict__ cf)
{
    int k = blockIdx.x * blockDim.x + threadIdx.x;
    if (k < HD) {
        float s = 0.f;
        for (int j = 0; j < 128; ++j) s += w1[(size_t)k * 128 + j] * w2[j];
        v[k] = s;
    }
    if (k == 0) {
        float c = 0.f;
        for (int j = 0; j < 128; ++j) c += b1[j] * w2[j];
        cf[0] = c;
    }
}

// ---------------- wsum[p] = sum_n leaky_relu(zp[p][n].v + c, 0.01) ----------------
__global__ void k_sem_w(const float* __restrict__ zp, const float* __restrict__ v,
                        const float* __restrict__ cf, float* __restrict__ wsum)
{
    const int wid  = (blockIdx.x * blockDim.x + threadIdx.x) >> 5;  // p*N + n
    const int lane = threadIdx.x & 31;
    if (wid >= P_PATHS * N_NODES) return;
    const float* __restrict__ z = zp + (size_t)wid * HD;
    float s = 0.f;
    for (int k = lane; k < HD; k += 32) s += z[k] * v[k];
    for (int off = 16; off; off >>= 1) s += __shfl_xor(s, off, 32);
    if (lane == 0) {
        float w = s + cf[0];
        w = w > 0.f ? w : 0.01f * w;
        atomicAdd(&wsum[wid / N_NODES], w);
    }
}

// ---------------- beta = softmax(wsum / N) ----------------
__global__ void k_beta(const float* __restrict__ wsum, float* __restrict__ beta)
{
    if (threadIdx.x == 0 && blockIdx.x == 0) {
        float w[P_PATHS], e[P_PATHS], m = -1e30f, s = 0.f;
        for (int p = 0; p < P_PATHS; ++p) { w[p] = wsum[p] * (1.f / N_NODES); m = fmaxf(m, w[p]); }
        for (int p = 0; p < P_PATHS; ++p) { e[p] = __expf(w[p] - m); s += e[p]; }
        for (int p = 0; p < P_PATHS; ++p) beta[p] = e[p] / s;
    }
}

// ---------------- out = (sum_p beta_p * zp) @ pred_w + pred_b (fp32 WMMA) ----------
// grid: N/16 blocks of one wave; each wave: 16 rows x 64 cols, K = HD
__global__ void k_final_gemm(const float* __restrict__ zp,   // [P,N,HD]
                             const float* __restrict__ beta,
                             const float* __restrict__ pw,   // [HD,OUT]
                             const float* __restrict__ pb,
                             float* __restrict__ out)        // [N,OUT]
{
    const int lane = threadIdx.x;
    const int lo = lane & 15, hi = lane >> 4;
    const int row0 = blockIdx.x * 16;
    const float b0 = beta[0], b1 = beta[1], b2 = beta[2];
    const float* __restrict__ z0 = zp + (size_t)(row0 + lo) * HD;
    const float* __restrict__ z1 = z0 + (size_t)N_NODES * HD;
    const float* __restrict__ z2 = z1 + (size_t)N_NODES * HD;

    v8f acc0 = {}, acc1 = {}, acc2 = {}, acc3 = {};
    for (int k = 0; k < HD; k += 4) {
        const int ka = k + 2 * hi;
        v2f a;
        a.x = b0 * z0[ka]     + b1 * z1[ka]     + b2 * z2[ka];
        a.y = b0 * z0[ka + 1] + b1 * z1[ka + 1] + b2 * z2[ka + 1];
        const float* __restrict__ w0 = pw + (size_t)ka * OUT_F + lo;
        const float* __restrict__ w1 = w0 + OUT_F;
        v2f b;
        b.x = w0[0];  b.y = w1[0];  acc0 = wmma4(a, b, acc0);
        b.x = w0[16]; b.y = w1[16]; acc1 = wmma4(a, b, acc1);
        b.x = w0[32]; b.y = w1[32]; acc2 = wmma4(a, b, acc2);
        b.x = w0[48]; b.y = w1[48]; acc3 = wmma4(a, b, acc3);
    }
    #pragma unroll
    for (int r = 0; r < 8; ++r) {
        const size_t ro = (size_t)(row0 + r + 8 * hi) * OUT_F + lo;
        out[ro +  0] = acc0[r] + pb[lo +  0];
        out[ro + 16] = acc1[r] + pb[lo + 16];
        out[ro + 32] = acc2[r] + pb[lo + 32];
        out[ro + 48] = acc3[r] + pb[lo + 48];
    }
}

// ---------------- atten[src,dst] += alpha_mean * beta[p] ----------------
__global__ void k_atten(const int* __restrict__ srcl, const int* __restrict__ dstl,
                        const float* __restrict__ amean, const float* __restrict__ beta,
                        float* __restrict__ atten)
{
    int idx = blockIdx.x * blockDim.x + threadIdx.x;     // P*E
    if (idx >= P_PATHS * EDGES) return;
    const int p = idx / EDGES;
    atomicAdd(&atten[(size_t)srcl[idx] * N_NODES + dstl[idx]], amean[idx] * beta[p]);
}

// =====================================================================
extern "C" void kernel_launch(void* const* d_in, const int* in_sizes, int n_in,
                              void* d_out, int out_size, void* d_ws, size_t ws_size,
                              hipStream_t stream)
{
    const float* h    = (const float*)d_in[0];
    const int*   esrc = (const int*)  d_in[1];
    const int*   edst = (const int*)  d_in[2];
    const float* fc   = (const float*)d_in[3];
    const float* al   = (const float*)d_in[4];
    const float* ar   = (const float*)d_in[5];
    const float* bias = (const float*)d_in[6];
    const float* w1   = (const float*)d_in[7];
    const float* b1   = (const float*)d_in[8];
    const float* w2   = (const float*)d_in[9];
    const float* pw   = (const float*)d_in[10];
    const float* pb   = (const float*)d_in[11];

    float* out   = (float*)d_out;                         // [N,OUT]
    float* atten = out + (size_t)N_NODES * OUT_F;         // [N,N]

    // workspace carve-up (floats)
    const size_t FEAT = (size_t)P_PATHS * N_NODES * HD;       // 12.58M
    const size_t EB   = (size_t)P_PATHS * EDGES * HEADS;      // 6.29M
    const size_t NH   = (size_t)P_PATHS * N_NODES * HEADS;    // 196K
    float*    ws    = (float*)d_ws;
    float*    feat  = ws;                 // [P,N,HD]
    float*    rst   = feat + FEAT;        // [P,N,HD] -> becomes zp after elu
    float*    ebuf  = rst + FEAT;         // [P,E,H]  e -> ee
    float*    el    = ebuf + EB;          // [P,N,H]
    float*    er    = el + NH;
    float*    denom = er + NH;
    unsigned* emax  = (unsigned*)(denom + NH);
    float*    amean = (float*)emax + NH;  // [P,E]
    float*    vvec  = amean + (size_t)P_PATHS * EDGES;  // [HD]
    float*    cf    = vvec + HD;          // [1]
    float*    wsum  = cf + 1;             // [P]
    float*    beta  = wsum + P_PATHS;     // [P]

    const int T = 256;
    // ---- per-call init (graph replays must not accumulate) ----
    k_fill_f32<<<(int)((FEAT + T - 1) / T), T, 0, stream>>>(rst, 0.f, (int)FEAT);
    k_fill_f32<<<(int)((NH + T - 1) / T), T, 0, stream>>>(denom, 0.f, (int)NH);
    k_fill_u32<<<(int)((NH + T - 1) / T), T, 0, stream>>>(emax, 0xFF800000u, (int)NH);
    k_fill_f32<<<1, 32, 0, stream>>>(wsum, 0.f, P_PATHS);
    const size_t NN = (size_t)N_NODES * N_NODES;
    k_fill_f32<<<(int)((NN + T - 1) / T), T, 0, stream>>>(atten, 0.f, (int)NN);

    // ---- GAT per path (batched over grid.z) ----
    k_gemm_feat<<<dim3(HD / 64, N_NODES / 16, P_PATHS), 32, 0, stream>>>(h, fc, feat);
    k_el_er<<<(int)((NH + T - 1) / T), T, 0, stream>>>(feat, al, ar, el, er);
    const size_t EBH = EB;
    k_edge_logits<<<(int)((EBH + T - 1) / T), T, 0, stream>>>(esrc, edst, el, er, ebuf, emax);
    k_edge_exp<<<(int)((EBH + T - 1) / T), T, 0, stream>>>(edst, emax, ebuf, denom);
    const size_t EW = (size_t)P_PATHS * EDGES * 32;   // one wave per edge
    k_edge_scatter<<<(int)((EW + T - 1) / T), T, 0, stream>>>(esrc, edst, ebuf, denom,
                                                              feat, rst, amean);
    k_bias_elu<<<(int)((FEAT + T - 1) / T), T, 0, stream>>>(rst, bias);

    // ---- semantic attention (linear-collapsed) ----
    k_sem_v<<<2, T, 0, stream>>>(w1, b1, w2, vvec, cf);
    const size_t SW = (size_t)P_PATHS * N_NODES * 32;
    k_sem_w<<<(int)((SW + T - 1) / T), T, 0, stream>>>(rst, vvec, cf, wsum);
    k_beta<<<1, 32, 0, stream>>>(wsum, beta);

    // ---- outputs ----
    k_final_gemm<<<N_NODES / 16, 32, 0, stream>>>(rst, beta, pw, pb, out);
    const size_t PE = (size_t)P_PATHS * EDGES;
    k_atten<<<(int)((PE + T - 1) / T), T, 0, stream>>>(esrc, edst, amean, beta, atten);
}